// RBF_gaussian_43422119362883
// MI455X (gfx1250) — compile-verified
//
#include <hip/hip_runtime.h>

typedef __attribute__((ext_vector_type(2))) float v2f;
typedef __attribute__((ext_vector_type(4))) float v4f;
typedef __attribute__((ext_vector_type(8))) float v8f;

#define IN_F 64
#define OUT_F 512

// ---------------------------------------------------------------------------
// Prep kernel (runs once, 512 lanes): per-center squared norm and 1/sigma^2.
//   ws[0   .. 511] = |c_o|^2
//   ws[512 ..1023] = exp(-2 * log_sigma_o)
// ---------------------------------------------------------------------------
__global__ __launch_bounds__(256) void rbf_prep_centers(
    const float* __restrict__ centers,     // (512, 64)
    const float* __restrict__ log_sigmas,  // (512,)
    float* __restrict__ ws)
{
    const int o = blockIdx.x * 256 + threadIdx.x;
    if (o >= OUT_F) return;
    const v4f* row = (const v4f*)(centers + (size_t)o * IN_F);
    float s = 0.f;
#pragma unroll
    for (int i = 0; i < IN_F / 4; ++i) {
        v4f c = row[i];
        s = fmaf(c.x, c.x, s);
        s = fmaf(c.y, c.y, s);
        s = fmaf(c.z, c.z, s);
        s = fmaf(c.w, c.w, s);
    }
    ws[o]         = s;
    ws[OUT_F + o] = __expf(-2.f * log_sigmas[o]);
}

// ---------------------------------------------------------------------------
// Main kernel: one wave computes a 16(M) x 64(N) tile of
//   out = exp(-max(|x|^2 - 2 x.c + |c|^2, 0) / sigma^2)
// Cross term via V_WMMA_F32_16X16X4_F32. Row norms accumulated from the A
// fragments in-register (2 fmas per K-step) and reduced with shfl_xor(16);
// center norms / inv-sigma^2 come precomputed from ws.
// ---------------------------------------------------------------------------
__global__ __launch_bounds__(256) void rbf_gaussian_wmma(
    const float* __restrict__ input,    // (N, 64) row-major
    const float* __restrict__ centers,  // (512, 64) row-major
    const float* __restrict__ ws,       // [csq(512) | invs(512)]
    float* __restrict__ out)            // (N, 512) row-major
{
    const int lane = threadIdx.x & 31;
    const int wave = threadIdx.x >> 5;
    const int half = lane >> 4;   // 0: lanes 0-15, 1: lanes 16-31
    const int l    = lane & 15;

    const int m0 = (blockIdx.x * 8 + wave) * 16;  // 16-row slab per wave
    const int n0 = blockIdx.y * 64;               // 64-col slab per block

    // A fragment: row (m0+l), K offset 2*half within each K=4 slice.
    const float* aptr = input   + (size_t)(m0 + l) * IN_F + 2 * half;
    // B fragment (centers^T has the symmetric lane layout): same pattern.
    const float* bptr = centers + (size_t)(n0 + l) * IN_F + 2 * half;

    v8f acc0 = {}, acc1 = {}, acc2 = {}, acc3 = {};
    float xp = 0.f;

#pragma unroll
    for (int s = 0; s < 16; ++s) {
        v2f a  = *(const v2f*)(aptr + 4 * s);
        v2f b0 = *(const v2f*)(bptr + 4 * s);
        v2f b1 = *(const v2f*)(bptr + 16 * IN_F + 4 * s);
        v2f b2 = *(const v2f*)(bptr + 32 * IN_F + 4 * s);
        v2f b3 = *(const v2f*)(bptr + 48 * IN_F + 4 * s);

        xp = fmaf(a.x, a.x, fmaf(a.y, a.y, xp));  // partial |x_row|^2

        acc0 = __builtin_amdgcn_wmma_f32_16x16x4_f32(false, a, false, b0, (short)0, acc0, false, false);
        acc1 = __builtin_amdgcn_wmma_f32_16x16x4_f32(false, a, false, b1, (short)0, acc1, false, false);
        acc2 = __builtin_amdgcn_wmma_f32_16x16x4_f32(false, a, false, b2, (short)0, acc2, false, false);
        acc3 = __builtin_amdgcn_wmma_f32_16x16x4_f32(false, a, false, b3, (short)0, acc3, false, false);
    }

    // finish row norm: lane (h,l) now holds |x_{m0+l}|^2
    xp += __shfl_xor(xp, 16, 32);

    // precomputed per-column terms (broadcast within each 16-lane half)
    const float cs0 = ws[n0      + l], is0 = ws[OUT_F + n0      + l];
    const float cs1 = ws[n0 + 16 + l], is1 = ws[OUT_F + n0 + 16 + l];
    const float cs2 = ws[n0 + 32 + l], is2 = ws[OUT_F + n0 + 32 + l];
    const float cs3 = ws[n0 + 48 + l], is3 = ws[OUT_F + n0 + 48 + l];

    // epilogue: C/D layout -> VGPR r holds row (r + 8*half), col (n0 + 16j + l)
    float* orow = out + (size_t)(m0 + 8 * half) * OUT_F + n0 + l;
#pragma unroll
    for (int r = 0; r < 8; ++r) {
        const float xs = __shfl(xp, r + 8 * half, 32);  // |x|^2 of this VGPR's row
        float* o = orow + (size_t)r * OUT_F;
        float d2;
        d2 = fmaxf(fmaf(-2.f, acc0[r], xs + cs0), 0.f); o[ 0] = __expf(-d2 * is0);
        d2 = fmaxf(fmaf(-2.f, acc1[r], xs + cs1), 0.f); o[16] = __expf(-d2 * is1);
        d2 = fmaxf(fmaf(-2.f, acc2[r], xs + cs2), 0.f); o[32] = __expf(-d2 * is2);
        d2 = fmaxf(fmaf(-2.f, acc3[r], xs + cs3), 0.f); o[48] = __expf(-d2 * is3);
    }
}

extern "C" void kernel_launch(void* const* d_in, const int* in_sizes, int n_in,
                              void* d_out, int out_size, void* d_ws, size_t ws_size,
                              hipStream_t stream) {
    const float* input      = (const float*)d_in[0];
    const float* centers    = (const float*)d_in[1];
    const float* log_sigmas = (const float*)d_in[2];
    float* ws  = (float*)d_ws;   // needs 1024 floats (4 KB)
    float* out = (float*)d_out;

    rbf_prep_centers<<<dim3(2), dim3(256), 0, stream>>>(centers, log_sigmas, ws);

    const int N = in_sizes[0] / IN_F;          // 262144
    dim3 grid(N / 128, OUT_F / 64);            // 8 waves/block * 16 rows = 128 rows/block
    rbf_gaussian_wmma<<<grid, dim3(256), 0, stream>>>(input, centers, ws, out);
}